// CostVolumeLayer_90683939487850
// MI455X (gfx1250) — compile-verified
//
#include <hip/hip_runtime.h>

typedef __attribute__((ext_vector_type(2))) float v2f;
typedef __attribute__((ext_vector_type(8))) float v8f;

#define B_   8
#define C_   128
#define H_   128
#define W_   256
#define HW_  (H_ * W_)
#define CHW_ (C_ * H_ * W_)
#define NSHIFT 9
#define KFRAG (C_ / 4)   // 32 chained K=4 WMMAs cover C=128

// One wave (32 lanes) handles one 16-pixel f1 strip (b, h, w0..w0+15) and all
// 81 shifts. 8 waves per block cover a 128-pixel row segment.
__global__ __launch_bounds__(256)
__attribute__((amdgpu_waves_per_eu(4, 8)))
void cost_volume_wmma_f32(
    const float* __restrict__ f1, const float* __restrict__ f2,
    float* __restrict__ out)
{
    __shared__ float band[8][NSHIFT * 16];   // per-wave 9x16 band staging

    const int lane = threadIdx.x & 31;
    const int wave = threadIdx.x >> 5;
    const int nl   = lane & 15;              // col-in-tile / row m selector
    const int hi   = lane >> 4;              // half-wave (K/M split)

    const int tw = blockIdx.x & 1;                 // which 128-wide half of W
    const int h  = (blockIdx.x >> 1) & (H_ - 1);
    const int b  = blockIdx.x >> 8;
    const int w0 = tw * 128 + wave * 16;

    // ---- A fragments: f1 strip, resident in VGPRs across all 9 row shifts.
    // Fully-unrolled constant indexing everywhere so SROA keeps the array in
    // registers (partial unroll previously forced it to scratch/LDS).
    // lane: m = nl, k = 4*kk + 2*hi + {0,1}
    const float* f1p = f1 + b * CHW_ + h * W_ + (w0 + nl) + (2 * hi) * HW_;
    v2f a[KFRAG];
#pragma unroll
    for (int kk = 0; kk < KFRAG; ++kk) {
        a[kk].x = f1p[(4 * kk + 0) * HW_];
        a[kk].y = f1p[(4 * kk + 1) * HW_];
    }

    // f2 column info (loop invariant): N tiles based at wb = w0-4 and wb+16
    const int wb   = w0 - 4;
    const int col0 = wb + nl;
    const int col1 = wb + 16 + nl;
    const bool v0w = (unsigned)col0 < (unsigned)W_;
    const bool v1w = (unsigned)col1 < (unsigned)W_;
    const int c0c  = col0 < 0 ? 0 : (col0 > W_ - 1 ? W_ - 1 : col0);
    const int c1c  = col1 < 0 ? 0 : (col1 > W_ - 1 ? W_ - 1 : col1);
    // Wave-uniform: does this wave's 32-col N strip stay fully inside [0, W)?
    const bool interior = (w0 != 0) && (w0 != W_ - 16);

    const float inv_c = 1.0f / (float)C_;

    for (int di = 0; di < NSHIFT; ++di) {
        const int hrow   = h + di - 4;                // block-uniform
        const bool rowok = (unsigned)hrow < (unsigned)H_;

        if (rowok) {
            const float* p0 = f2 + b * CHW_ + hrow * W_ + c0c + (2 * hi) * HW_;
            const float* p1 = f2 + b * CHW_ + hrow * W_ + c1c + (2 * hi) * HW_;
            v8f acc0 = {};
            v8f acc1 = {};
            if (interior) {
                // Fast path: no boundary masking needed.
#pragma unroll
                for (int kk = 0; kk < KFRAG; ++kk) {
                    v2f b0, b1;
                    b0.x = p0[(4 * kk + 0) * HW_];
                    b0.y = p0[(4 * kk + 1) * HW_];
                    b1.x = p1[(4 * kk + 0) * HW_];
                    b1.y = p1[(4 * kk + 1) * HW_];
                    acc0 = __builtin_amdgcn_wmma_f32_16x16x4_f32(
                        false, a[kk], false, b0, (short)0, acc0, false, false);
                    acc1 = __builtin_amdgcn_wmma_f32_16x16x4_f32(
                        false, a[kk], false, b1, (short)0, acc1, false, false);
                }
            } else {
                // Edge path: clamp-loaded values masked to zero out of range.
#pragma unroll
                for (int kk = 0; kk < KFRAG; ++kk) {
                    float t00 = p0[(4 * kk + 0) * HW_];
                    float t01 = p0[(4 * kk + 1) * HW_];
                    float t10 = p1[(4 * kk + 0) * HW_];
                    float t11 = p1[(4 * kk + 1) * HW_];
                    v2f b0, b1;
                    b0.x = v0w ? t00 : 0.0f;
                    b0.y = v0w ? t01 : 0.0f;
                    b1.x = v1w ? t10 : 0.0f;
                    b1.y = v1w ? t11 : 0.0f;
                    acc0 = __builtin_amdgcn_wmma_f32_16x16x4_f32(
                        false, a[kk], false, b0, (short)0, acc0, false, false);
                    acc1 = __builtin_amdgcn_wmma_f32_16x16x4_f32(
                        false, a[kk], false, b1, (short)0, acc1, false, false);
                }
            }
            // Scatter band entries (dj = n - m in [0,8]) into LDS.
            // D layout: lane, VGPR r -> (m = r + 8*hi, n = nl [+16 tile 1])
#pragma unroll
            for (int r = 0; r < 8; ++r) {
                const int m   = r + 8 * hi;
                const int dj0 = nl - m;              // tile 0
                if (dj0 >= 0 && dj0 < NSHIFT)
                    band[wave][dj0 * 16 + m] = acc0[r];
                const int dj1 = nl + 16 - m;         // tile 1 (always >= 1)
                if (dj1 < NSHIFT)
                    band[wave][dj1 * 16 + m] = acc1[r];
            }
        }
        __syncthreads();

        // Store 9 dj-rows of 16 contiguous floats; two rows per iteration
        // (half-wave each). Zero rows for out-of-range hrow.
        float* outp = out + (((b * 81) + di * NSHIFT) * H_ + h) * W_ + w0 + nl;
#pragma unroll
        for (int p = 0; p < 5; ++p) {
            const int dj = 2 * p + hi;
            if (dj < NSHIFT) {
                const float v =
                    rowok ? band[wave][dj * 16 + nl] * inv_c : 0.0f;
                outp[dj * HW_] = v;
            }
        }
        __syncthreads();
    }
}

extern "C" void kernel_launch(void* const* d_in, const int* in_sizes, int n_in,
                              void* d_out, int out_size, void* d_ws, size_t ws_size,
                              hipStream_t stream) {
    const float* f1 = (const float*)d_in[0];
    const float* f2 = (const float*)d_in[1];
    float* out = (float*)d_out;
    (void)in_sizes; (void)n_in; (void)out_size; (void)d_ws; (void)ws_size;

    dim3 grid(B_ * H_ * (W_ / 128));   // 8 * 128 * 2 = 2048 blocks
    dim3 block(256);                   // 8 wave32 per block, one 16-px tile each
    hipLaunchKernelGGL(cost_volume_wmma_f32, grid, block, 0, stream,
                       f1, f2, out);
}